// Euclidean_79671643341035
// MI455X (gfx1250) — compile-verified
//
#include <hip/hip_runtime.h>

typedef __attribute__((ext_vector_type(2))) float v2f;
typedef __attribute__((ext_vector_type(8))) float v8f;

#define B_ROWS   4096
#define D_DIM    64
#define K_COLS   50000
#define NTILE_N  25                 // 16-wide N-tiles per wave strip
#define N_STRIPS (3125 / NTILE_N)   // 50000/16 = 3125 N-tiles -> 125 strips
#define M_GROUPS (B_ROWS / 32)      // two 16-row M-tiles per wave -> 128

// ---- prologue: row norms of x -------------------------------------------
__global__ __launch_bounds__(256) void euclid_x2_kernel(const float* __restrict__ x,
                                                        float* __restrict__ x2) {
  int i = blockIdx.x * blockDim.x + threadIdx.x;
  if (i < B_ROWS) {
    const float4* p = (const float4*)(x + (size_t)i * D_DIM);
    float s = 0.f;
#pragma unroll
    for (int k = 0; k < D_DIM / 4; ++k) {
      float4 v = p[k];
      s += v.x * v.x + v.y * v.y + v.z * v.z + v.w * v.w;
    }
    x2[i] = s;
  }
}

// ---- prologue: column norms of w ----------------------------------------
__global__ __launch_bounds__(256) void euclid_w2_kernel(const float* __restrict__ w,
                                                        float* __restrict__ w2) {
  int j = blockIdx.x * blockDim.x + threadIdx.x;
  if (j < K_COLS) {
    float s = 0.f;
#pragma unroll
    for (int k = 0; k < D_DIM; ++k) {
      float v = w[(size_t)k * K_COLS + j];
      s += v * v;
    }
    w2[j] = s;
  }
}

// ---- main: D = (-2x) @ w + (x2 + w2), then sqrt(max(D,eps)) -------------
__global__ __launch_bounds__(256) void euclid_wmma_kernel(const float* __restrict__ x,
                                                          const float* __restrict__ w,
                                                          const float* __restrict__ x2,
                                                          const float* __restrict__ w2,
                                                          float* __restrict__ out) {
  const int lane = threadIdx.x & 31;
  const int wave = blockIdx.x * (blockDim.x >> 5) + (threadIdx.x >> 5);
  const int half = lane >> 4;   // 0: lanes 0-15, 1: lanes 16-31
  const int lr   = lane & 15;

  const int m_group  = wave / N_STRIPS;
  const int n_strip  = wave % N_STRIPS;
  const int row0     = m_group * 32;
  const int col_base = n_strip * (NTILE_N * 16);

  // A matrix (16x4 f32 per WMMA): lane holds K = 2*half, 2*half+1 of each
  // 4-wide k-block for its row.  Pre-scale by -2 so WMMA accumulates -2*x@w.
  v2f a[2][16];
#pragma unroll
  for (int t = 0; t < 2; ++t) {
    const float* xr = x + (size_t)(row0 + 16 * t + lr) * D_DIM + 2 * half;
#pragma unroll
    for (int kb = 0; kb < 16; ++kb) {
      a[t][kb].x = -2.0f * xr[4 * kb + 0];
      a[t][kb].y = -2.0f * xr[4 * kb + 1];
    }
  }

  // x2 values matching the C/D layout: VGPR v <-> row (v + 8*half)
  float x2v[2][8];
#pragma unroll
  for (int t = 0; t < 2; ++t)
#pragma unroll
    for (int v = 0; v < 8; ++v)
      x2v[t][v] = x2[row0 + 16 * t + 8 * half + v];

  for (int nt = 0; nt < NTILE_N; ++nt) {
    const int col = col_base + nt * 16 + lr;
    const float w2c = w2[col];

    v8f c0, c1;
#pragma unroll
    for (int v = 0; v < 8; ++v) {
      c0[v] = x2v[0][v] + w2c;
      c1[v] = x2v[1][v] + w2c;
    }

    // B matrix (4x16 f32): lane holds K = 2*half, 2*half+1 for its column.
    const float* wc = w + (size_t)(2 * half) * K_COLS + col;
#pragma unroll
    for (int kb = 0; kb < 16; ++kb) {
      v2f b;
      b.x = wc[(size_t)(4 * kb + 0) * K_COLS];
      b.y = wc[(size_t)(4 * kb + 1) * K_COLS];
      c0 = __builtin_amdgcn_wmma_f32_16x16x4_f32(false, a[0][kb], false, b,
                                                 (short)0, c0, false, false);
      c1 = __builtin_amdgcn_wmma_f32_16x16x4_f32(false, a[1][kb], false, b,
                                                 (short)0, c1, false, false);
    }

#pragma unroll
    for (int v = 0; v < 8; ++v) {
      float d0 = fmaxf(c0[v], 1e-12f);
      float d1 = fmaxf(c1[v], 1e-12f);
      out[(size_t)(row0 + 8 * half + v) * K_COLS + col]      = sqrtf(d0);
      out[(size_t)(row0 + 16 + 8 * half + v) * K_COLS + col] = sqrtf(d1);
    }
  }
}

extern "C" void kernel_launch(void* const* d_in, const int* in_sizes, int n_in,
                              void* d_out, int out_size, void* d_ws, size_t ws_size,
                              hipStream_t stream) {
  (void)in_sizes; (void)n_in; (void)out_size; (void)ws_size;
  const float* x = (const float*)d_in[0];     // [4096, 64]
  const float* w = (const float*)d_in[1];     // [64, 50000]
  float* out = (float*)d_out;                 // [4096, 50000]

  float* x2 = (float*)d_ws;                   // 4096 floats
  float* w2 = x2 + B_ROWS;                    // 50000 floats

  euclid_x2_kernel<<<(B_ROWS + 255) / 256, 256, 0, stream>>>(x, x2);
  euclid_w2_kernel<<<(K_COLS + 255) / 256, 256, 0, stream>>>(w, w2);

  const int total_waves = M_GROUPS * N_STRIPS;     // 128 * 125 = 16000
  const int blocks = total_waves / 8;              // 8 waves / 256-thread block
  euclid_wmma_kernel<<<blocks, 256, 0, stream>>>(x, w, x2, w2, out);
}